// TemporalConvDownScaleAttention_44710609552115
// MI455X (gfx1250) — compile-verified
//
#include <hip/hip_runtime.h>

// out[b,t1,c,hw] = 0.25 * sum_i x[b,4*t1+i,c,hw] + 0.25 * sum_i pos_table[i,hw]
// Pure HBM-bandwidth problem (~160 MiB @ 23.3 TB/s ~= 7us). The K=4 temporal
// reduction maps exactly onto V_WMMA_F32_16X16X4_F32: A = const 0.25 (layout-
// invariant), B = 4 time samples per column. C is a loop-invariant zero so the
// backend never has to re-materialize the accumulator; the pos bias is added
// with one v_add_f32 after the lane-select.

typedef __attribute__((ext_vector_type(2))) float v2f;
typedef __attribute__((ext_vector_type(8))) float v8f;

#define HW       4096      // h*w
#define TSTRIDE  262144    // c*h*w = 64*4096 floats between consecutive t slices

__global__ __launch_bounds__(256)
void tmean_wmma_kernel(const float* __restrict__ x,
                       const float* __restrict__ pos,
                       float* __restrict__ out)
{
    __shared__ float pm[HW];   // 16 KiB: pos_mean[hw]

    const int tid = threadIdx.x;

    // Cooperative precompute of pos_mean into LDS (pos_table is 4 x 4096).
    for (int j = tid; j < HW; j += 256) {
        pm[j] = 0.25f * (pos[j] + pos[j + HW] + pos[j + 2 * HW] + pos[j + 3 * HW]);
    }
    __syncthreads();

    // One block per output row: row = B*64 + c, with B = b*4 + t1 in [0,32).
    const int row = blockIdx.x;
    const int B   = row >> 6;
    const int c   = row & 63;

    // x row base for i=0: ((b*16 + 4*t1)*64 + c)*4096 = (256*B + c)*4096
    const float* xr   = x + (size_t)(B * 256 + c) * HW;
    float*       orow = out + (size_t)row * HW;

    const int  lane = tid & 31;
    const int  wave = tid >> 5;
    const bool lo   = (lane < 16);

    // Lanes 0-15 carry time slices {0,1}; lanes 16-31 carry {2,3}.
    const float* pA = xr + (size_t)(lo ? 0 : 2) * TSTRIDE;
    const float* pB = xr + (size_t)(lo ? 1 : 3) * TSTRIDE;

    // A-matrix: every element 0.25 -> layout-independent.
    const v2f a = { 0.25f, 0.25f };
    // Loop-invariant zero accumulator (SRC2 can be inline 0 / shared VGPRs).
    const v8f czero = {};

    // Each wave owns a 512-element strip; 16 iterations x 32 outputs.
    for (int it = 0; it < 16; ++it) {
        const int base = wave * 512 + it * 32;
        const int hw0  = base + (lane & 15);

        // B-matrix tiles: column n = lane%16, two K-slices per lane.
        v2f b0 = { pA[hw0],      pB[hw0]      };   // tile0: hw [base, base+16)
        v2f b1 = { pA[hw0 + 16], pB[hw0 + 16] };   // tile1: hw [base+16, base+32)

        const float pmv = pm[base + lane];

        v8f d0 = __builtin_amdgcn_wmma_f32_16x16x4_f32(
            false, a, false, b0, (short)0, czero, false, false);
        v8f d1 = __builtin_amdgcn_wmma_f32_16x16x4_f32(
            false, a, false, b1, (short)0, czero, false, false);

        // Lanes 0-15: tile0 D[0, lane]; lanes 16-31: tile1 D[8, lane-16].
        // Together: 32 contiguous floats -> one 128B coalesced store.
        orow[base + lane] = (lo ? d0[0] : d1[0]) + pmv;
    }
}

extern "C" void kernel_launch(void* const* d_in, const int* in_sizes, int n_in,
                              void* d_out, int out_size, void* d_ws, size_t ws_size,
                              hipStream_t stream) {
    const float* x   = (const float*)d_in[0];   // (8,16,64,64,64) f32
    const float* pos = (const float*)d_in[1];   // (4, 4096) f32
    float* out       = (float*)d_out;           // (8,4,64,64,64) f32

    // 2048 output rows (B*c), 256 threads = 8 wave32 per block.
    tmean_wmma_kernel<<<2048, 256, 0, stream>>>(x, pos, out);
}